// Lorec_67353677136189
// MI455X (gfx1250) — compile-verified
//
#include <hip/hip_runtime.h>
#include <hip/hip_bf16.h>

// ---- problem constants ----
#define BS        8
#define SEQ       2048
#define IN_DIM    4096
#define OUT_DIM   4096
#define R         16
#define CTR_OUT   256
#define CTR_HID   60
#define FINAL_DIM 16
#define LN_EPS    1e-5f
#define SCALING   2.0f   // ALPHA / R = 32/16

typedef __attribute__((ext_vector_type(2))) float v2f;
typedef __attribute__((ext_vector_type(8))) float v8f;

// ---------------------------------------------------------------------------
// Kernel 1: gate[b][16] = softmax(MLP(LayerNorm(ctr_hidden_states[b])))
// Tiny serial work: one thread per batch (8 active threads total).
// ---------------------------------------------------------------------------
__global__ void lorec_gate_kernel(const float* __restrict__ ctr,
                                  const float* __restrict__ gamma,
                                  const float* __restrict__ beta,
                                  const float* __restrict__ W1,
                                  const float* __restrict__ b1,
                                  const float* __restrict__ W2,
                                  const float* __restrict__ b2,
                                  float* __restrict__ gate) {
    __shared__ float hsh[BS][CTR_OUT];
    const int b = threadIdx.x;
    if (b >= BS) return;
    const float* xr = ctr + b * CTR_OUT;

    float s = 0.f, s2 = 0.f;
    for (int j = 0; j < CTR_OUT; ++j) { float v = xr[j]; s += v; s2 += v * v; }
    const float mu   = s * (1.0f / CTR_OUT);
    const float var  = s2 * (1.0f / CTR_OUT) - mu * mu;
    const float rstd = rsqrtf(var + LN_EPS);

    float* h = hsh[b];
    for (int j = 0; j < CTR_OUT; ++j)
        h[j] = (xr[j] - mu) * rstd * gamma[j] + beta[j];

    float h1[CTR_HID];
    for (int k = 0; k < CTR_HID; ++k) {
        float a = b1[k];
        const float* w = W1 + k * CTR_OUT;
        for (int j = 0; j < CTR_OUT; ++j) a += h[j] * w[j];
        h1[k] = a > 0.f ? a : 0.f;
    }

    float lg[FINAL_DIM];
    float mx = -3.0e38f;
    for (int m = 0; m < FINAL_DIM; ++m) {
        float a = b2[m];
        const float* w = W2 + m * CTR_HID;
        for (int k = 0; k < CTR_HID; ++k) a += h1[k] * w[k];
        lg[m] = a;
        mx = a > mx ? a : mx;
    }
    float se = 0.f;
    for (int m = 0; m < FINAL_DIM; ++m) { lg[m] = __expf(lg[m] - mx); se += lg[m]; }
    const float inv = 1.0f / se;
    for (int m = 0; m < FINAL_DIM; ++m) gate[b * FINAL_DIM + m] = lg[m] * inv;
}

// ---------------------------------------------------------------------------
// Kernel 2: per-batch adapters.
//   At[b][r][i] = sum_f gate[b][f] * Wa[r*IN_DIM  + i][f]   (row-major [16][4096])
//   Bm[b][r][o] = sum_f gate[b][f] * Wb[r*OUT_DIM + o][f]   (row-major [16][4096])
// One thread per Wa/Wb row; read the 16 weights once, reuse for all 8 batches.
// ---------------------------------------------------------------------------
__global__ void lorec_adapter_kernel(const float* __restrict__ Wa,
                                     const float* __restrict__ Wb,
                                     const float* __restrict__ gate,
                                     float* __restrict__ At,
                                     float* __restrict__ Bm) {
    const int j = blockIdx.x * blockDim.x + threadIdx.x;   // 0 .. R*IN_DIM-1
    if (j >= R * IN_DIM) return;
    const float* wa = Wa + (size_t)j * FINAL_DIM;
    const float* wb = Wb + (size_t)j * FINAL_DIM;
    float wav[FINAL_DIM], wbv[FINAL_DIM];
#pragma unroll
    for (int f = 0; f < FINAL_DIM; ++f) { wav[f] = wa[f]; wbv[f] = wb[f]; }
#pragma unroll
    for (int b = 0; b < BS; ++b) {
        const float* g = gate + b * FINAL_DIM;
        float sa = 0.f, sb = 0.f;
#pragma unroll
        for (int f = 0; f < FINAL_DIM; ++f) { sa += g[f] * wav[f]; sb += g[f] * wbv[f]; }
        At[(size_t)b * R * IN_DIM + j]  = sa;
        Bm[(size_t)b * R * OUT_DIM + j] = sb;
    }
}

// ---------------------------------------------------------------------------
// Kernel 3: main chained low-rank GEMM with V_WMMA_F32_16X16X4_F32.
// grid: BS * (SEQ/16) blocks, 128 threads = 4 waves (wave32).
//  Pass 1: xa(16x16) = x_tile(16x4096) @ At_b^T(4096x16), K split over 4 waves,
//          256 chained wmma per wave; partials reduced through LDS.
//  Pass 2: out_tile(16x4096) = xa @ Bm_b(16x4096), each wave owns 1024 columns
//          = 64 tiles x 4 chained K=4 wmma; store out * SCALING.
// ---------------------------------------------------------------------------
__global__ __launch_bounds__(128) void lorec_main_kernel(
        const float* __restrict__ x,
        const float* __restrict__ At,
        const float* __restrict__ Bm,
        float* __restrict__ out) {
    __shared__ float part[4][256];   // per-wave 16x16 partial xa tiles (row-major [M][r])

    const int tid  = threadIdx.x;
    const int wave = tid >> 5;
    const int lane = tid & 31;
    const int n    = lane & 15;   // column / row-in-tile index (0..15)
    const int hi   = lane >> 4;   // K-half selector for A/B fragments

    const int b  = blockIdx.x >> 7;      // batch (SEQ/16 = 128 tiles per batch)
    const int s0 = (blockIdx.x & 127) * 16;

    const float* xb = x  + (size_t)b * SEQ * IN_DIM;
    const float* Ab = At + (size_t)b * R * IN_DIM;
    const float* Bb = Bm + (size_t)b * R * OUT_DIM;

    // ---- Pass 1: accumulate xa partial over this wave's K range ----
    v8f acc = {};
    const int KW = IN_DIM / 4;           // 1024 columns per wave
    const int k0 = wave * KW;
    // A-operand (x tile, 16x4 f32): lanes 0-15 hold K={k,k+1}, lanes 16-31 K={k+2,k+3}, M = lane&15
    const float* xp = xb + (size_t)(s0 + n) * IN_DIM + k0 + hi * 2;
    // B-operand (At, 4x16 f32): lane column = n, v0/v1 = At[b][n][k + 2*hi + {0,1}]
    const float* ap = Ab + (size_t)n * IN_DIM + k0 + hi * 2;
#pragma unroll 4
    for (int k = 0; k < KW; k += 4) {
        v2f afrag = *(const v2f*)(xp + k);
        v2f bfrag = *(const v2f*)(ap + k);
        acc = __builtin_amdgcn_wmma_f32_16x16x4_f32(
            /*neg_a=*/false, afrag, /*neg_b=*/false, bfrag,
            /*c_mod=*/(short)0, acc, /*reuse_a=*/false, /*reuse_b=*/false);
    }

    // Spill partial to LDS in plain [M][r] matrix form (D layout: M = v + 8*hi, N = n)
#pragma unroll
    for (int v = 0; v < 8; ++v)
        part[wave][(v + 8 * hi) * 16 + n] = acc[v];
    __syncthreads();

    // Reduce the 4 partials into part[0]
    for (int i = tid; i < 256; i += 128) {
        float sum = part[0][i] + part[1][i] + part[2][i] + part[3][i];
        part[0][i] = sum;
    }
    __syncthreads();

    // ---- Pass 2: out_tile = xa @ Bm_b, this wave handles 1024 output columns ----
    // xa as A-operand fragments for K chunks kr*4: v0/v1 = xa[M=lane&15][kr*4 + 2*hi + {0,1}]
    v2f xaf[4];
    const float* xam = &part[0][0];
#pragma unroll
    for (int kr = 0; kr < 4; ++kr) {
        xaf[kr].x = xam[n * 16 + kr * 4 + hi * 2];
        xaf[kr].y = xam[n * 16 + kr * 4 + hi * 2 + 1];
    }

    float* outb = out + (size_t)b * SEQ * OUT_DIM + (size_t)s0 * OUT_DIM;
#pragma unroll 2
    for (int t = 0; t < 64; ++t) {
        const int o0 = wave * 1024 + t * 16;
        v8f d = {};
#pragma unroll
        for (int kr = 0; kr < 4; ++kr) {
            // B-operand (Bm tile, 4x16): lane column n; rows kr*4 + 2*hi + {0,1}
            const float* bp = Bb + (size_t)(kr * 4 + hi * 2) * OUT_DIM + o0 + n;
            v2f bfrag;
            bfrag.x = bp[0];
            bfrag.y = bp[OUT_DIM];
            d = __builtin_amdgcn_wmma_f32_16x16x4_f32(
                false, xaf[kr], false, bfrag, (short)0, d, false, false);
        }
#pragma unroll
        for (int v = 0; v < 8; ++v)
            outb[(size_t)(v + 8 * hi) * OUT_DIM + o0 + n] = d[v] * SCALING;
    }
}

// ---------------------------------------------------------------------------
extern "C" void kernel_launch(void* const* d_in, const int* in_sizes, int n_in,
                              void* d_out, int out_size, void* d_ws, size_t ws_size,
                              hipStream_t stream) {
    const float* x     = (const float*)d_in[0];
    const float* ctr   = (const float*)d_in[1];
    const float* gamma = (const float*)d_in[2];
    const float* beta  = (const float*)d_in[3];
    const float* W1    = (const float*)d_in[4];
    const float* b1    = (const float*)d_in[5];
    const float* W2    = (const float*)d_in[6];
    const float* b2    = (const float*)d_in[7];
    const float* Wa    = (const float*)d_in[8];
    const float* Wb    = (const float*)d_in[9];
    float* out = (float*)d_out;

    // workspace layout (fp32): gate[8*16] | At[8*16*4096] | Bm[8*16*4096]  (~4.2 MB)
    float* ws   = (float*)d_ws;
    float* gate = ws;
    float* At   = ws + 128;
    float* Bm   = At + (size_t)BS * R * IN_DIM;

    lorec_gate_kernel<<<1, 32, 0, stream>>>(ctr, gamma, beta, W1, b1, W2, b2, gate);
    lorec_adapter_kernel<<<(R * IN_DIM + 255) / 256, 256, 0, stream>>>(Wa, Wb, gate, At, Bm);
    lorec_main_kernel<<<BS * (SEQ / 16), 128, 0, stream>>>(x, At, Bm, out);
}